// Block_19945828122792
// MI455X (gfx1250) — compile-verified
//
#include <hip/hip_runtime.h>
#include <math.h>

#define B_      2
#define L_      1024
#define DMODEL  1024
#define DINNER  2048
#define DSTATE  16
#define DCONV   4
#define DTRANK  64
#define DFF     4096
#define CCH     16        // scan chunks
#define LC      (L_ / CCH)

typedef __attribute__((ext_vector_type(2))) float v2f;
typedef __attribute__((ext_vector_type(8))) float v8f;
typedef int vi4 __attribute__((vector_size(16)));
typedef __attribute__((address_space(1))) vi4* as1_vi4p;
typedef __attribute__((address_space(3))) vi4* as3_vi4p;

__device__ __forceinline__ float siluf(float v) { return v / (1.0f + __expf(-v)); }

// ---------- gfx1250 async global->LDS copy (ASYNCcnt-tracked, no VGPR staging) ----------
#if defined(__has_builtin)
#if __has_builtin(__builtin_amdgcn_global_load_async_to_lds_b128)
#define ASYNC_BUILTIN 1
#endif
#if __has_builtin(__builtin_amdgcn_s_wait_asynccnt)
#define WAITASYNC_BUILTIN 1
#endif
#endif

__device__ __forceinline__ void async_b128(const float* g, float* l) {
#if defined(ASYNC_BUILTIN)
  as1_vi4p gp = (as1_vi4p)(unsigned long long)(uintptr_t)g;
  as3_vi4p lp = (as3_vi4p)(unsigned int)(uintptr_t)l;
  __builtin_amdgcn_global_load_async_to_lds_b128(gp, lp, 0, 0);
#else
  asm volatile("global_load_async_to_lds_b128 %0, %1, off"
               :: "v"((unsigned)(unsigned long long)l), "v"(g)
               : "memory");
#endif
}

__device__ __forceinline__ void wait_async0() {
#if defined(WAITASYNC_BUILTIN)
  __builtin_amdgcn_s_wait_asynccnt(0);
#else
  asm volatile("s_wait_asynccnt 0x0" ::: "memory");
#endif
}
__device__ __forceinline__ void wait_async3() {
#if defined(WAITASYNC_BUILTIN)
  __builtin_amdgcn_s_wait_asynccnt(3);
#else
  asm volatile("s_wait_asynccnt 0x3" ::: "memory");
#endif
}

// ---------------- LayerNorm over last dim (1024) ----------------
__global__ __launch_bounds__(256)
void ln_kernel(const float* __restrict__ x, const float* __restrict__ w,
               const float* __restrict__ b, float* __restrict__ out) {
  const int row = blockIdx.x;
  const float* xr = x + (size_t)row * DMODEL;
  float* orow = out + (size_t)row * DMODEL;
  float v[4]; float s = 0.f, s2 = 0.f;
#pragma unroll
  for (int i = 0; i < 4; ++i) {
    v[i] = xr[threadIdx.x + 256 * i];
    s += v[i]; s2 += v[i] * v[i];
  }
  __shared__ float rs[256], rq[256];
  rs[threadIdx.x] = s; rq[threadIdx.x] = s2;
  __syncthreads();
  for (int off = 128; off > 0; off >>= 1) {
    if (threadIdx.x < off) {
      rs[threadIdx.x] += rs[threadIdx.x + off];
      rq[threadIdx.x] += rq[threadIdx.x + off];
    }
    __syncthreads();
  }
  const float mu   = rs[0] * (1.0f / DMODEL);
  const float var  = rq[0] * (1.0f / DMODEL) - mu * mu;
  const float rstd = rsqrtf(var + 1e-5f);
#pragma unroll
  for (int i = 0; i < 4; ++i) {
    const int c = threadIdx.x + 256 * i;
    orow[c] = (v[i] - mu) * rstd * w[c] + b[c];
  }
}

// ---------------- WMMA f32 GEMM:  C = act(A(MxK) * W(NxK)^T + bias) (+residual) ----------------
// act: 0 = none, 1 = relu, 2 = softplus
// Block tile 64(M) x 128(N) x 16(K); 8 wave32s; each wave: one 16x64 strip (4 x v8f accums).
// Full-N path: double-buffered GLOBAL_LOAD_ASYNC_TO_LDS_B128 pipeline (ASYNCcnt).
#define BM 64
#define BN 128
#define BK 16

__global__ __launch_bounds__(256)
void gemm_wmma(const float* __restrict__ A, int lda,
               const float* __restrict__ W, int ldw,
               float* __restrict__ C, int ldc,
               const float* __restrict__ bias,
               const float* __restrict__ residual,
               int M, int N, int K, int act) {
  __shared__ float As[2][BM][BK + 1];
  __shared__ float Ws[2][BN][BK + 1];
  const int m0 = blockIdx.y * BM;
  const int n0 = blockIdx.x * BN;
  const int t    = threadIdx.x;
  const int lrow = t >> 2;          // 0..63
  const int lcol = (t & 3) * 4;     // 0,4,8,12
  const int wid  = t >> 5;          // wave id 0..7
  const int lane = t & 31;
  const int half = lane >> 4;       // 0/1
  const int l16  = lane & 15;
  const int wr   = (wid & 3) * 16;  // wave M-subtile offset in block tile
  const int wc   = (wid >> 2) * 64; // wave N-offset (four 16-wide subtiles)

  v8f acc[4] = {};

  const float* Arow  = A + (size_t)(m0 + lrow) * lda + lcol;
  const float* Wrow0 = W + (size_t)(n0 + lrow) * ldw + lcol;
  const float* Wrow1 = W + (size_t)(n0 + 64 + lrow) * ldw + lcol;

  const bool full = ((N & (BN - 1)) == 0);   // uniform: all tiles in-bounds

  if (full) {
    const int niter = K / BK;
    // prologue: stage tile 0 into buffer 0 (3 async B128 per thread -> wave ASYNCcnt += 3)
    async_b128(Arow,  &As[0][lrow][lcol]);
    async_b128(Wrow0, &Ws[0][lrow][lcol]);
    async_b128(Wrow1, &Ws[0][64 + lrow][lcol]);
    for (int i = 0; i < niter; ++i) {
      const int cur = i & 1;
      if (i + 1 < niter) {
        const int nk = (i + 1) * BK;
        const int nb = cur ^ 1;      // consumed in the previous iteration; barrier below protects it
        async_b128(Arow  + nk, &As[nb][lrow][lcol]);
        async_b128(Wrow0 + nk, &Ws[nb][lrow][lcol]);
        async_b128(Wrow1 + nk, &Ws[nb][64 + lrow][lcol]);
        wait_async3();               // cur tile done (in-order), next tile still in flight
      } else {
        wait_async0();
      }
      __syncthreads();
#pragma unroll
      for (int kk = 0; kk < BK; kk += 4) {
        v2f a;
        a.x = As[cur][wr + l16][kk + 2 * half];
        a.y = As[cur][wr + l16][kk + 2 * half + 1];
#pragma unroll
        for (int j = 0; j < 4; ++j) {
          v2f b;
          b.x = Ws[cur][wc + 16 * j + l16][kk + 2 * half];
          b.y = Ws[cur][wc + 16 * j + l16][kk + 2 * half + 1];
          acc[j] = __builtin_amdgcn_wmma_f32_16x16x4_f32(false, a, false, b,
                                                         (short)0, acc[j], false, false);
        }
      }
      __syncthreads();
    }
  } else {
    // -------- guarded path (partial N tiles, e.g. N = 96) --------
    const bool wv0 = (n0 + lrow) < N;
    const bool wv1 = (n0 + 64 + lrow) < N;
    for (int k0 = 0; k0 < K; k0 += BK) {
      {
        const float4 av = *(const float4*)(Arow + k0);
        As[0][lrow][lcol + 0] = av.x; As[0][lrow][lcol + 1] = av.y;
        As[0][lrow][lcol + 2] = av.z; As[0][lrow][lcol + 3] = av.w;
      }
      if (wv0) {
        const float4 wvv = *(const float4*)(Wrow0 + k0);
        Ws[0][lrow][lcol + 0] = wvv.x; Ws[0][lrow][lcol + 1] = wvv.y;
        Ws[0][lrow][lcol + 2] = wvv.z; Ws[0][lrow][lcol + 3] = wvv.w;
      } else {
        Ws[0][lrow][lcol + 0] = 0.f; Ws[0][lrow][lcol + 1] = 0.f;
        Ws[0][lrow][lcol + 2] = 0.f; Ws[0][lrow][lcol + 3] = 0.f;
      }
      if (wv1) {
        const float4 wvv = *(const float4*)(Wrow1 + k0);
        Ws[0][64 + lrow][lcol + 0] = wvv.x; Ws[0][64 + lrow][lcol + 1] = wvv.y;
        Ws[0][64 + lrow][lcol + 2] = wvv.z; Ws[0][64 + lrow][lcol + 3] = wvv.w;
      } else {
        Ws[0][64 + lrow][lcol + 0] = 0.f; Ws[0][64 + lrow][lcol + 1] = 0.f;
        Ws[0][64 + lrow][lcol + 2] = 0.f; Ws[0][64 + lrow][lcol + 3] = 0.f;
      }
      __syncthreads();
      if (k0 + BK < K) __builtin_prefetch(Arow + k0 + BK, 0, 3);
#pragma unroll
      for (int kk = 0; kk < BK; kk += 4) {
        v2f a;
        a.x = As[0][wr + l16][kk + 2 * half];
        a.y = As[0][wr + l16][kk + 2 * half + 1];
#pragma unroll
        for (int j = 0; j < 4; ++j) {
          v2f b;
          b.x = Ws[0][wc + 16 * j + l16][kk + 2 * half];
          b.y = Ws[0][wc + 16 * j + l16][kk + 2 * half + 1];
          acc[j] = __builtin_amdgcn_wmma_f32_16x16x4_f32(false, a, false, b,
                                                         (short)0, acc[j], false, false);
        }
      }
      __syncthreads();
    }
  }

  // C/D layout: VGPR v -> M = v + 8*half ; N = lane&15
#pragma unroll
  for (int j = 0; j < 4; ++j) {
    const int n = n0 + wc + 16 * j + l16;
    if (n < N) {
#pragma unroll
      for (int v = 0; v < 8; ++v) {
        const int m = m0 + wr + v + 8 * half;
        float val = acc[j][v];
        if (bias) val += bias[n];
        if (act == 1)      val = fmaxf(val, 0.f);
        else if (act == 2) val = (val > 20.f) ? val : log1pf(__expf(val));
        if (residual) val += residual[(size_t)m * ldc + n];
        C[(size_t)m * ldc + n] = val;
      }
    }
  }
}

// ---------------- depthwise causal conv (k=4) + bias + SiLU ----------------
__global__ __launch_bounds__(256)
void conv_silu(const float* __restrict__ xz, const float* __restrict__ cw,
               const float* __restrict__ cb, float* __restrict__ u) {
  const int idx = blockIdx.x * 256 + threadIdx.x;    // over B*L*DINNER
  const int d = idx & (DINNER - 1);
  const int l = (idx >> 11) & (L_ - 1);
  const int b = idx >> 21;
  float acc = cb[d];
#pragma unroll
  for (int j = 0; j < DCONV; ++j) {
    const int ls = l - (DCONV - 1) + j;
    if (ls >= 0)
      acc += cw[d * DCONV + j] * xz[(size_t)(b * L_ + ls) * (2 * DINNER) + d];
  }
  u[idx] = siluf(acc);
}

// ======== chunked-parallel selective scan (h_t = dA_t*h_{t-1} + dt_t*B_t*u_t) ========
// Buffers P,Q,Hin laid out [b][chunk][s][DINNER] so d is the fast (coalesced) axis.
__device__ __forceinline__ size_t pq_idx(int b, int c, int s, int d) {
  return (((size_t)(b * CCH + c) * DSTATE) + s) * DINNER + d;
}

// Phase 1: per-chunk local scan with h=0 -> chunk transition (P = prod dA, Q = local end state)
__global__ __launch_bounds__(256)
void scan_p1(const float* __restrict__ xdb, const float* __restrict__ dtb,
             const float* __restrict__ u, const float* __restrict__ A_log,
             float* __restrict__ Pb, float* __restrict__ Qb) {
  const int blk = blockIdx.x;                // ((b*8 + dc)<<4) | c
  const int c  = blk & (CCH - 1);
  const int dc = (blk >> 4) & 7;
  const int b  = blk >> 7;
  const int d  = dc * 256 + threadIdx.x;
  float Ac[DSTATE];
#pragma unroll
  for (int s = 0; s < DSTATE; ++s) Ac[s] = -__expf(A_log[d * DSTATE + s]);
  float pr[DSTATE], st[DSTATE];
#pragma unroll
  for (int s = 0; s < DSTATE; ++s) { pr[s] = 1.f; st[s] = 0.f; }
  __shared__ float BC[32];
  for (int l = c * LC; l < (c + 1) * LC; ++l) {
    const size_t tok = (size_t)(b * L_ + l);
    if (threadIdx.x < 32) BC[threadIdx.x] = xdb[tok * 96 + DTRANK + threadIdx.x];
    __syncthreads();
    const float dt = dtb[tok * DINNER + d];
    const float uu = u[tok * DINNER + d];
#pragma unroll
    for (int s = 0; s < DSTATE; ++s) {
      const float dA = __expf(dt * Ac[s]);
      st[s] = dA * st[s] + dt * BC[s] * uu;
      pr[s] *= dA;
    }
    __syncthreads();
  }
#pragma unroll
  for (int s = 0; s < DSTATE; ++s) {
    Pb[pq_idx(b, c, s, d)] = pr[s];
    Qb[pq_idx(b, c, s, d)] = st[s];
  }
}

// Phase 2: sequential combine over the 16 chunks -> incoming state per chunk
__global__ __launch_bounds__(256)
void scan_p2(const float* __restrict__ Pb, const float* __restrict__ Qb,
             float* __restrict__ Hin) {
  const int tid = blockIdx.x * 256 + threadIdx.x;   // b*DINNER + d
  const int b = tid >> 11;
  const int d = tid & (DINNER - 1);
  float hcur[DSTATE];
#pragma unroll
  for (int s = 0; s < DSTATE; ++s) hcur[s] = 0.f;
  for (int c = 0; c < CCH; ++c) {
#pragma unroll
    for (int s = 0; s < DSTATE; ++s) {
      const size_t ix = pq_idx(b, c, s, d);
      Hin[ix] = hcur[s];
      hcur[s] = Pb[ix] * hcur[s] + Qb[ix];
    }
  }
}

// Phase 3: re-run chunk scans seeded with Hin; emit y fused with D-skip + SiLU(z) gate
__global__ __launch_bounds__(256)
void scan_p3(const float* __restrict__ xz, const float* __restrict__ u,
             const float* __restrict__ xdb, const float* __restrict__ dtb,
             const float* __restrict__ A_log, const float* __restrict__ Dp,
             const float* __restrict__ Hin, float* __restrict__ y) {
  const int blk = blockIdx.x;
  const int c  = blk & (CCH - 1);
  const int dc = (blk >> 4) & 7;
  const int b  = blk >> 7;
  const int d  = dc * 256 + threadIdx.x;
  float Ac[DSTATE];
#pragma unroll
  for (int s = 0; s < DSTATE; ++s) Ac[s] = -__expf(A_log[d * DSTATE + s]);
  const float Dd = Dp[d];
  float st[DSTATE];
#pragma unroll
  for (int s = 0; s < DSTATE; ++s) st[s] = Hin[pq_idx(b, c, s, d)];
  __shared__ float BC[32];
  for (int l = c * LC; l < (c + 1) * LC; ++l) {
    const size_t tok = (size_t)(b * L_ + l);
    if (threadIdx.x < 32) BC[threadIdx.x] = xdb[tok * 96 + DTRANK + threadIdx.x];
    __syncthreads();
    const float dt = dtb[tok * DINNER + d];
    const float uu = u[tok * DINNER + d];
    const float z  = xz[tok * (2 * DINNER) + DINNER + d];
    float accy = 0.f;
#pragma unroll
    for (int s = 0; s < DSTATE; ++s) {
      const float dA = __expf(dt * Ac[s]);
      st[s] = dA * st[s] + dt * BC[s] * uu;
      accy += st[s] * BC[16 + s];
    }
    y[tok * DINNER + d] = (accy + uu * Dd) * siluf(z);
    __syncthreads();
  }
}

// ---------------- host launcher ----------------
extern "C" void kernel_launch(void* const* d_in, const int* in_sizes, int n_in,
                              void* d_out, int out_size, void* d_ws, size_t ws_size,
                              hipStream_t stream) {
  const float* x      = (const float*)d_in[0];
  const float* ln1_w  = (const float*)d_in[1];
  const float* ln1_b  = (const float*)d_in[2];
  const float* W_in   = (const float*)d_in[3];
  const float* conv_w = (const float*)d_in[4];
  const float* conv_b = (const float*)d_in[5];
  const float* W_xprj = (const float*)d_in[6];
  const float* W_dt   = (const float*)d_in[7];
  const float* b_dt   = (const float*)d_in[8];
  const float* A_log  = (const float*)d_in[9];
  const float* Dp     = (const float*)d_in[10];
  const float* W_out  = (const float*)d_in[11];
  const float* ln2_w  = (const float*)d_in[12];
  const float* ln2_b  = (const float*)d_in[13];
  const float* W1     = (const float*)d_in[14];
  const float* b1     = (const float*)d_in[15];
  const float* W2     = (const float*)d_in[16];
  const float* b2     = (const float*)d_in[17];
  float* out = (float*)d_out;

  const size_t NT = (size_t)B_ * L_;   // 2048 tokens
  const size_t PQ = (size_t)B_ * CCH * DSTATE * DINNER;  // 1M floats per buffer
  float* ws  = (float*)d_ws;
  float* h    = ws;                 // NT*1024
  float* xz   = h   + NT * DMODEL;  // NT*4096
  float* u    = xz  + NT * 4096;    // NT*2048
  float* xdb  = u   + NT * DINNER;  // NT*96
  float* dt   = xdb + NT * 96;      // NT*2048
  float* y    = dt  + NT * DINNER;  // NT*2048
  float* x2   = y   + NT * DINNER;  // NT*1024
  float* h2   = x2  + NT * DMODEL;  // NT*1024
  float* Pb   = h2  + NT * DMODEL;  // PQ
  float* Qb   = Pb  + PQ;           // PQ
  float* Hin  = Qb  + PQ;           // PQ
  float* ffm  = xz;                 // alias: xz is dead after the scan

  const dim3 blk(256);

  // 1) ln1
  ln_kernel<<<dim3((unsigned)NT), blk, 0, stream>>>(x, ln1_w, ln1_b, h);
  // 2) xz = h * W_in^T   (2048 x 4096 x 1024)
  gemm_wmma<<<dim3(4096 / BN, (unsigned)NT / BM), blk, 0, stream>>>(
      h, DMODEL, W_in, DMODEL, xz, 4096, nullptr, nullptr,
      (int)NT, 2 * DINNER, DMODEL, 0);
  // 3) depthwise conv + SiLU -> u
  conv_silu<<<dim3((unsigned)(NT * DINNER / 256)), blk, 0, stream>>>(
      xz, conv_w, conv_b, u);
  // 4) x_db = u * W_xproj^T   (2048 x 96 x 2048)
  gemm_wmma<<<dim3((96 + BN - 1) / BN, (unsigned)NT / BM), blk, 0, stream>>>(
      u, DINNER, W_xprj, DINNER, xdb, 96, nullptr, nullptr,
      (int)NT, DTRANK + 2 * DSTATE, DINNER, 0);
  // 5) dt = softplus(x_db[:, :64] * W_dt^T + b_dt)   (2048 x 2048 x 64)
  gemm_wmma<<<dim3(DINNER / BN, (unsigned)NT / BM), blk, 0, stream>>>(
      xdb, 96, W_dt, DTRANK, dt, DINNER, b_dt, nullptr,
      (int)NT, DINNER, DTRANK, 2);
  // 6) chunk-parallel selective scan (fused D-skip + SiLU(z) gate in phase 3)
  scan_p1<<<dim3(B_ * 8 * CCH), blk, 0, stream>>>(xdb, dt, u, A_log, Pb, Qb);
  scan_p2<<<dim3(B_ * DINNER / 256), blk, 0, stream>>>(Pb, Qb, Hin);
  scan_p3<<<dim3(B_ * 8 * CCH), blk, 0, stream>>>(xz, u, xdb, dt, A_log, Dp, Hin, y);
  // 7) x2 = x + y * W_out^T   (2048 x 1024 x 2048)
  gemm_wmma<<<dim3(DMODEL / BN, (unsigned)NT / BM), blk, 0, stream>>>(
      y, DINNER, W_out, DINNER, x2, DMODEL, nullptr, x,
      (int)NT, DMODEL, DINNER, 0);
  // 8) ln2
  ln_kernel<<<dim3((unsigned)NT), blk, 0, stream>>>(x2, ln2_w, ln2_b, h2);
  // 9) ffm = relu(h2 * W1^T + b1)   (2048 x 4096 x 1024)
  gemm_wmma<<<dim3(DFF / BN, (unsigned)NT / BM), blk, 0, stream>>>(
      h2, DMODEL, W1, DMODEL, ffm, DFF, b1, nullptr,
      (int)NT, DFF, DMODEL, 1);
  // 10) out = x2 + ffm * W2^T + b2   (2048 x 1024 x 4096)
  gemm_wmma<<<dim3(DMODEL / BN, (unsigned)NT / BM), blk, 0, stream>>>(
      ffm, DFF, W2, DFF, out, DMODEL, b2, x2,
      (int)NT, DMODEL, DFF, 0);
}